// PredictAndOptimize_27650999452171
// MI455X (gfx1250) — compile-verified
//
#include <hip/hip_runtime.h>
#include <math.h>

// ---------------------------------------------------------------------------
// PredictAndOptimize for MI455X (gfx1250, wave32)
//   Kernel 1: MLP (128->64->32->1) with V_WMMA_F32_16X16X4_F32 fp32 matrix ops
//   Kernel 2: batched simplex QP via projected gradient (one wave32 / sample,
//             Q rows resident in VGPRs, Michelot exact simplex projection)
// ---------------------------------------------------------------------------

typedef float v8f __attribute__((ext_vector_type(8)));
typedef float v2f __attribute__((ext_vector_type(2)));

#define BATCH   4096
#define KDIM    50
#define FDIM    128
#define H1DIM   64
#define H2DIM   32
#define NROWS   (BATCH * KDIM)        // 204800, divisible by 16
#define GAMMA_C 5.0f
#define LN_EPS_C 1e-5f
#define PGD_ITERS 300
#define POW_ITERS 80

// ===========================================================================
// Kernel 1: fused MLP. One wave (32 threads) computes a 16-row slab.
// ===========================================================================
__global__ __launch_bounds__(32)
void mlp_wmma_kernel(const float* __restrict__ X,
                     const float* __restrict__ W1, const float* __restrict__ b1,
                     const float* __restrict__ g1, const float* __restrict__ be1,
                     const float* __restrict__ W2, const float* __restrict__ b2,
                     const float* __restrict__ g2, const float* __restrict__ be2,
                     const float* __restrict__ W3, const float* __restrict__ b3,
                     float* __restrict__ mu_out)
{
    __shared__ float Xs[16 * FDIM];    // 8 KB input slab
    __shared__ float H1[16 * H1DIM];   // 4 KB
    __shared__ float H2[16 * H2DIM];   // 2 KB
    __shared__ float mrow[16];
    __shared__ float srow[16];

    const int lane = threadIdx.x;          // 0..31
    const int half = lane >> 4;            // 0: lanes 0-15, 1: lanes 16-31
    const int m    = lane & 15;            // row (A/C) or column (B) index
    const long r0  = (long)blockIdx.x * 16;

    // ---- stage X slab (16 rows x 128 f32, contiguous) into LDS, b128 loads
    {
        const float4* Xg4 = (const float4*)(X + r0 * FDIM);
        float4* Xs4 = (float4*)Xs;
#pragma unroll
        for (int i = 0; i < 16; ++i)
            Xs4[lane + 32 * i] = Xg4[lane + 32 * i];
    }
    __syncthreads();

    const v8f zero8 = {0.f, 0.f, 0.f, 0.f, 0.f, 0.f, 0.f, 0.f};

    // ---------------- Layer 1: (16x128) @ (128x64) via WMMA f32 16x16x4 ----
    v8f acc1[4];
#pragma unroll
    for (int n = 0; n < 4; ++n) acc1[n] = zero8;

#pragma unroll
    for (int kc = 0; kc < FDIM / 4; ++kc) {
        const int k0 = kc * 4;
        // A fragment: lane half 0 -> K=k0,k0+1 ; half 1 -> K=k0+2,k0+3
        v2f a;
        a.x = Xs[m * FDIM + k0 + 2 * half];
        a.y = Xs[m * FDIM + k0 + 2 * half + 1];
#pragma unroll
        for (int n = 0; n < 4; ++n) {
            // B fragment: lane(mod 16) = N column, VGPR/half = K row
            v2f bf;
            bf.x = W1[(k0 + 2 * half) * H1DIM + 16 * n + m];
            bf.y = W1[(k0 + 2 * half + 1) * H1DIM + 16 * n + m];
            acc1[n] = __builtin_amdgcn_wmma_f32_16x16x4_f32(
                false, a, false, bf, (short)0, acc1[n], false, false);
        }
    }

    // spill C tiles (+bias) to LDS: vgpr j holds row j (lanes 0-15) / j+8
#pragma unroll
    for (int n = 0; n < 4; ++n) {
#pragma unroll
        for (int j = 0; j < 8; ++j) {
            const int row = j + 8 * half;
            const int col = 16 * n + m;
            H1[row * H1DIM + col] = acc1[n][j] + b1[col];
        }
    }
    __syncthreads();

    // ---- LayerNorm1 + ReLU (biased variance, matches reference) ----------
    if (lane < 16) {
        float s = 0.f, ss = 0.f;
#pragma unroll
        for (int c = 0; c < H1DIM; ++c) {
            const float x = H1[lane * H1DIM + c];
            s += x; ss += x * x;
        }
        const float mean = s * (1.f / H1DIM);
        const float var  = ss * (1.f / H1DIM) - mean * mean;
        mrow[lane] = mean;
        srow[lane] = 1.f / sqrtf(var + LN_EPS_C);
    }
    __syncthreads();
#pragma unroll
    for (int i = 0; i < 32; ++i) {           // 1024 elements / 32 lanes
        const int idx = lane + 32 * i;
        const int row = idx >> 6, col = idx & 63;
        const float x = (H1[idx] - mrow[row]) * srow[row] * g1[col] + be1[col];
        H1[idx] = fmaxf(x, 0.f);
    }
    __syncthreads();

    // ---------------- Layer 2: (16x64) @ (64x32) via WMMA ------------------
    v8f acc2[2];
#pragma unroll
    for (int n = 0; n < 2; ++n) acc2[n] = zero8;

#pragma unroll
    for (int kc = 0; kc < H1DIM / 4; ++kc) {
        const int k0 = kc * 4;
        v2f a;
        a.x = H1[m * H1DIM + k0 + 2 * half];
        a.y = H1[m * H1DIM + k0 + 2 * half + 1];
#pragma unroll
        for (int n = 0; n < 2; ++n) {
            v2f bf;
            bf.x = W2[(k0 + 2 * half) * H2DIM + 16 * n + m];
            bf.y = W2[(k0 + 2 * half + 1) * H2DIM + 16 * n + m];
            acc2[n] = __builtin_amdgcn_wmma_f32_16x16x4_f32(
                false, a, false, bf, (short)0, acc2[n], false, false);
        }
    }
#pragma unroll
    for (int n = 0; n < 2; ++n) {
#pragma unroll
        for (int j = 0; j < 8; ++j) {
            const int row = j + 8 * half;
            const int col = 16 * n + m;
            H2[row * H2DIM + col] = acc2[n][j] + b2[col];
        }
    }
    __syncthreads();

    // ---- LayerNorm2 + ReLU ------------------------------------------------
    if (lane < 16) {
        float s = 0.f, ss = 0.f;
#pragma unroll
        for (int c = 0; c < H2DIM; ++c) {
            const float x = H2[lane * H2DIM + c];
            s += x; ss += x * x;
        }
        const float mean = s * (1.f / H2DIM);
        const float var  = ss * (1.f / H2DIM) - mean * mean;
        mrow[lane] = mean;
        srow[lane] = 1.f / sqrtf(var + LN_EPS_C);
    }
    __syncthreads();
#pragma unroll
    for (int i = 0; i < 16; ++i) {           // 512 elements / 32 lanes
        const int idx = lane + 32 * i;
        const int row = idx >> 5, col = idx & 31;
        const float x = (H2[idx] - mrow[row]) * srow[row] * g2[col] + be2[col];
        H2[idx] = fmaxf(x, 0.f);
    }
    __syncthreads();

    // ---------------- Layer 3: (16x32) @ (32x1) ----------------------------
    if (lane < 16) {
        float acc = b3[0];
#pragma unroll
        for (int k = 0; k < H2DIM; ++k)
            acc = fmaf(H2[lane * H2DIM + k], W3[k], acc);
        mu_out[r0 + lane] = acc;
    }
}

// ===========================================================================
// Kernel 2: batched QP. One wave32 per sample, 8 samples per 256-thread block.
// Q rows live in VGPRs; w broadcast with lane shuffles; Michelot projection.
// ===========================================================================
__device__ __forceinline__ float wave_sum32(float x) {
#pragma unroll
    for (int o = 16; o > 0; o >>= 1) x += __shfl_xor(x, o, 32);
    return x;
}

__global__ __launch_bounds__(256)
void qp_pgd_kernel(const float* __restrict__ Sigma,
                   const float* __restrict__ mu_all,
                   float* __restrict__ w_out)
{
    const int lane = threadIdx.x & 31;
    const int wave = threadIdx.x >> 5;
    const int b    = blockIdx.x * 8 + wave;

    const float* __restrict__ S = Sigma + (long)b * (KDIM * KDIM);

    const int i0 = lane;            // component 0..31 (always < 50)
    const int i1 = lane + 32;       // component 32..63 (valid if < 50)
    const bool ok1 = (i1 < KDIM);

    // ---- build my two rows of Q = gamma*(sym(Sigma) + 1e-6 I) in VGPRs ----
    float Q0[KDIM], Q1[KDIM];
#pragma unroll
    for (int j = 0; j < KDIM; ++j) {
        Q0[j] = GAMMA_C * (0.5f * (S[i0 * KDIM + j] + S[j * KDIM + i0]) +
                           (j == i0 ? 1e-6f : 0.f));
        Q1[j] = ok1 ? GAMMA_C * (0.5f * (S[i1 * KDIM + j] + S[j * KDIM + i1]) +
                                 (j == i1 ? 1e-6f : 0.f))
                    : 0.f;
    }

    // matvec: (y0,y1) = Q @ x, x distributed 2 components per lane
#define MATVEC(x0, x1, y0, y1)                                   \
    do {                                                         \
        y0 = 0.f; y1 = 0.f;                                      \
        _Pragma("unroll")                                        \
        for (int j = 0; j < 32; ++j) {                           \
            const float xj = __shfl((x0), j, 32);                \
            y0 = fmaf(Q0[j], xj, y0);                            \
            y1 = fmaf(Q1[j], xj, y1);                            \
        }                                                        \
        _Pragma("unroll")                                        \
        for (int j = 32; j < KDIM; ++j) {                        \
            const float xj = __shfl((x1), j - 32, 32);           \
            y0 = fmaf(Q0[j], xj, y0);                            \
            y1 = fmaf(Q1[j], xj, y1);                            \
        }                                                        \
    } while (0)

    // ---- power iteration for lambda_max (step size eta = 1/lambda) --------
    float v0 = 1.f + 0.001f * (float)i0;
    float v1 = ok1 ? 1.f + 0.001f * (float)i1 : 0.f;
    float lam = 1.f;
    for (int it = 0; it < POW_ITERS; ++it) {
        float u0, u1;
        MATVEC(v0, v1, u0, u1);
        if (!ok1) u1 = 0.f;
        const float nrm = sqrtf(wave_sum32(u0 * u0 + u1 * u1));
        lam = nrm;
        const float inv = 1.f / nrm;
        v0 = u0 * inv;
        v1 = u1 * inv;
    }
    const float eta = 1.f / (lam + 1e-8f);

    // ---- load mu (written by kernel 1 into d_out's second region) ---------
    const float mu0 = mu_all[b * KDIM + i0];
    const float mu1 = ok1 ? mu_all[b * KDIM + i1] : 0.f;

    // ---- projected gradient descent ---------------------------------------
    float w0 = 1.f / (float)KDIM;
    float w1 = ok1 ? 1.f / (float)KDIM : 0.f;

    for (int it = 0; it < PGD_ITERS; ++it) {
        float g0, g1;
        MATVEC(w0, w1, g0, g1);
        g0 -= mu0;
        g1 -= mu1;
        const float a0 = w0 - eta * g0;
        const float a1 = ok1 ? (w1 - eta * g1) : -1e30f;  // sentinel: never active

        // Michelot: exact projection onto {w>=0, sum w = 1} without sorting.
        float tau;
        {
            const float s_all = wave_sum32(a0 + (ok1 ? a1 : 0.f));
            tau = (s_all - 1.f) / (float)KDIM;
        }
        int cnt_prev = KDIM;
        for (int mi = 0; mi < KDIM; ++mi) {   // wave-uniform loop + break
            const float s = wave_sum32((a0 > tau ? a0 : 0.f) +
                                       (a1 > tau ? a1 : 0.f));
            const float c = wave_sum32((a0 > tau ? 1.f : 0.f) +
                                       (a1 > tau ? 1.f : 0.f));
            const int cnt = (int)(c + 0.5f);
            tau = (s - 1.f) / c;
            if (cnt == cnt_prev) break;       // support stable => converged
            cnt_prev = cnt;
        }
        w0 = fmaxf(a0 - tau, 0.f);
        w1 = ok1 ? fmaxf(a1 - tau, 0.f) : 0.f;
    }

    // ---- write w_star ------------------------------------------------------
    w_out[b * KDIM + i0] = w0;
    if (ok1) w_out[b * KDIM + i1] = w1;
#undef MATVEC
}

// ===========================================================================
// launcher
// ===========================================================================
extern "C" void kernel_launch(void* const* d_in, const int* in_sizes, int n_in,
                              void* d_out, int out_size, void* d_ws, size_t ws_size,
                              hipStream_t stream) {
    const float* X   = (const float*)d_in[0];
    const float* Sig = (const float*)d_in[1];
    const float* W1  = (const float*)d_in[2];
    const float* b1  = (const float*)d_in[3];
    const float* g1  = (const float*)d_in[4];
    const float* be1 = (const float*)d_in[5];
    const float* W2  = (const float*)d_in[6];
    const float* b2  = (const float*)d_in[7];
    const float* g2  = (const float*)d_in[8];
    const float* be2 = (const float*)d_in[9];
    const float* W3  = (const float*)d_in[10];
    const float* b3  = (const float*)d_in[11];

    float* w_star = (float*)d_out;                  // (4096, 50)
    float* mu_hat = (float*)d_out + NROWS;          // (4096, 50)

    // Kernel 1: 204800 rows / 16 per wave-block
    mlp_wmma_kernel<<<NROWS / 16, 32, 0, stream>>>(
        X, W1, b1, g1, be1, W2, b2, g2, be2, W3, b3, mu_hat);

    // Kernel 2: 8 samples (waves) per 256-thread block
    qp_pgd_kernel<<<BATCH / 8, 256, 0, stream>>>(Sig, mu_hat, w_star);
}